// GNN_50517405335826
// MI455X (gfx1250) — compile-verified
//
#include <hip/hip_runtime.h>
#include <math.h>

// Problem constants (from reference)
#define NNODES 50000
#define NEDGES 600000
#define NGRAPH 8
#define DFEAT  128
#define HID1   128
#define HID2   64
#define DTGT   8

typedef __attribute__((ext_vector_type(2))) float v2f;
typedef __attribute__((ext_vector_type(8))) float v8f;

// ---- helpers -------------------------------------------------------------

// Order-preserving float<->uint encoding so atomicMax(uint) == float max.
__device__ __forceinline__ unsigned f2o(float f) {
    unsigned u = __float_as_uint(f);
    return (u & 0x80000000u) ? ~u : (u | 0x80000000u);
}
__device__ __forceinline__ float o2f(unsigned u) {
    return (u & 0x80000000u) ? __uint_as_float(u & 0x7FFFFFFFu)
                             : __uint_as_float(~u);
}
__device__ __forceinline__ float eluf(float v) {
    return v > 0.0f ? v : expm1f(v);
}

// ---- small-buffer init ---------------------------------------------------
// small layout (uint words):
// [0..7]   colmax1   [8..15] colmax2
// [16..23] colsum1(f)[24..31] colsum2(f)
// [32..95] segmax0   [96..159] segmax1   [160..223] segmax2
__global__ void init_small(unsigned* __restrict__ sm) {
    int t = threadIdx.x;
    if (t < 224) sm[t] = 0u;   // encoded 0u == -inf sentinel; 0u is also 0.0f for sums
}

__global__ void zero_f(float* __restrict__ p, int n) {
    int i = blockIdx.x * blockDim.x + threadIdx.x;
    if (i < n) p[i] = 0.0f;
}

// ---- edge scatter-add: agg[dst] += x[src], dim floats per row ------------
__global__ void scatter_add(const float* __restrict__ X,
                            const int* __restrict__ src,
                            const int* __restrict__ dst,
                            float* __restrict__ AGG,
                            int nEdges, int dim) {
    int perEdge = dim >> 2;                 // float4 chunks
    int t = blockIdx.x * blockDim.x + threadIdx.x;
    if (t >= nEdges * perEdge) return;
    int e = t / perEdge;
    int q = t - e * perEdge;
    int s = src[e];
    int d = dst[e];
    const float4 v = ((const float4*)(X + (size_t)s * dim))[q];
    float* o = AGG + (size_t)d * dim + q * 4;
    atomicAdd(o + 0, v.x);
    atomicAdd(o + 1, v.y);
    atomicAdd(o + 2, v.z);
    atomicAdd(o + 3, v.w);
}

// ---- WMMA f32 GEMM + fused BN/ELU epilogue -------------------------------
// Y[NNODES,NOUT] = elu(((X+AGG)[NNODES,K] @ W[K,NOUT] + B)*S + T)
// One wave computes MT vertically-stacked 16x16 tiles (same 16-col strip),
// so each B (weight) fragment load feeds MT V_WMMA_F32_16X16X4_F32 ops.
template <int K, int NOUT, int MT>
__global__ void conv_wmma(const float* __restrict__ X,
                          const float* __restrict__ AGG,
                          const float* __restrict__ W,
                          const float* __restrict__ B,
                          const float* __restrict__ S,
                          const float* __restrict__ T,
                          float* __restrict__ Y) {
    constexpr int TN      = NOUT / 16;           // N-tiles (compile-time)
    constexpr int NGROUPS = (NNODES / 16) / MT;  // M-groups (3125/MT, exact)

    int wave = (blockIdx.x * blockDim.x + threadIdx.x) >> 5;
    int lane = threadIdx.x & 31;
    if (wave >= NGROUPS * TN) return;            // wave-uniform guard
    int gm = wave / TN;                          // strength-reduced (TN is 2^k)
    int tn = wave - gm * TN;

    int half = lane >> 4;                        // 0: K+0/K+1   1: K+2/K+3
    int l    = lane & 15;
    int colB = tn * 16 + l;                      // B/C/D: N = lane&15

    const float* xr[MT];
    const float* ar[MT];
#pragma unroll
    for (int t = 0; t < MT; ++t) {
        int rowA = (gm * MT + t) * 16 + l;       // A: M = lane&15 (both halves)
        xr[t] = X   + (size_t)rowA * K;
        ar[t] = AGG + (size_t)rowA * K;
    }

    v8f acc[MT];
#pragma unroll
    for (int t = 0; t < MT; ++t) acc[t] = (v8f){};

#pragma unroll 4
    for (int k = 0; k < K; k += 4) {
        int ka = k + half * 2;
        v2f b;
        b.x = W[(size_t)(ka)     * NOUT + colB];
        b.y = W[(size_t)(ka + 1) * NOUT + colB];
#pragma unroll
        for (int t = 0; t < MT; ++t) {
            v2f a;
            a.x = xr[t][ka]     + ar[t][ka];
            a.y = xr[t][ka + 1] + ar[t][ka + 1];
            acc[t] = __builtin_amdgcn_wmma_f32_16x16x4_f32(
                /*neg_a=*/false, a, /*neg_b=*/false, b,
                /*c_mod=*/(short)0, acc[t], /*reuse_a=*/false, /*reuse_b=*/false);
        }
    }

    // Epilogue: C/D layout — VGPR r: lanes0-15 M=r, lanes16-31 M=r+8; N=lane&15
    int col = tn * 16 + l;
    float bb = B[col], ss = S[col], tt = T[col];
#pragma unroll
    for (int t = 0; t < MT; ++t) {
#pragma unroll
        for (int r = 0; r < 8; ++r) {
            int row = (gm * MT + t) * 16 + r + half * 8;
            float v = (acc[t][r] + bb) * ss + tt;
            Y[(size_t)row * NOUT + col] = eluf(v);
        }
    }
}

// ---- DT=8 score head: z = elu((x@W + b)*s + t), + segmax/colmax atomics --
__global__ void lin_head(const float* __restrict__ X, int K,
                         const float* __restrict__ W,
                         const float* __restrict__ B,
                         const float* __restrict__ S,
                         const float* __restrict__ T,
                         const int* __restrict__ batch,
                         float* __restrict__ Zout,
                         unsigned* __restrict__ colmax,   // nullable
                         unsigned* __restrict__ segmax) {
    int idx = blockIdx.x * blockDim.x + threadIdx.x;
    if (idx >= NNODES * DTGT) return;
    int n = idx >> 3;
    int c = idx & 7;
    const float* xr = X + (size_t)n * K;
    float acc = 0.0f;
#pragma unroll 8
    for (int k = 0; k < K; ++k) acc = fmaf(xr[k], W[k * DTGT + c], acc);
    float v = eluf((acc + B[c]) * S[c] + T[c]);
    Zout[idx] = v;
    unsigned enc = f2o(v);
    if (colmax) atomicMax(&colmax[c], enc);
    atomicMax(&segmax[(size_t)batch[n] * DTGT + c], enc);
}

// ---- column-wise sum of exp(z - colmax) with block reduction -------------
__global__ void col_expsum(const float* __restrict__ Zl,
                           const unsigned* __restrict__ colmax,
                           float* __restrict__ colsum) {
    __shared__ float sh[DTGT];
    if (threadIdx.x < DTGT) sh[threadIdx.x] = 0.0f;
    __syncthreads();
    int idx = blockIdx.x * blockDim.x + threadIdx.x;
    if (idx < NNODES * DTGT) {
        int c = idx & 7;
        atomicAdd(&sh[c], expf(Zl[idx] - o2f(colmax[c])));
    }
    __syncthreads();
    if (threadIdx.x < DTGT) atomicAdd(&colsum[threadIdx.x], sh[threadIdx.x]);
}

// ---- Z += softmax(z1)*z1 + softmax(z2)*z2 --------------------------------
__global__ void z_update(float* __restrict__ Zacc,
                         const float* __restrict__ z1,
                         const unsigned* __restrict__ cm1,
                         const float* __restrict__ cs1,
                         const float* __restrict__ z2,
                         const unsigned* __restrict__ cm2,
                         const float* __restrict__ cs2) {
    int idx = blockIdx.x * blockDim.x + threadIdx.x;
    if (idx >= NNODES * DTGT) return;
    int c = idx & 7;
    float a = z1[idx];
    float b = z2[idx];
    float r = Zacc[idx];
    r += expf(a - o2f(cm1[c])) / cs1[c] * a;
    r += expf(b - o2f(cm2[c])) / cs2[c] * b;
    Zacc[idx] = r;
}

// ---- out[g,c] = segmax0 + segmax1 + segmax2 ------------------------------
__global__ void out_finalize(const unsigned* __restrict__ seg,
                             float* __restrict__ out) {
    int i = threadIdx.x;
    if (i < NGRAPH * DTGT)
        out[i] = o2f(seg[i]) + o2f(seg[64 + i]) + o2f(seg[128 + i]);
}

// ---- host launcher -------------------------------------------------------
extern "C" void kernel_launch(void* const* d_in, const int* in_sizes, int n_in,
                              void* d_out, int out_size, void* d_ws, size_t ws_size,
                              hipStream_t stream) {
    const float* x      = (const float*)d_in[0];
    const int*   eidx   = (const int*)d_in[1];
    const int*   batch  = (const int*)d_in[2];
    const float* lin0_w = (const float*)d_in[3];
    const float* lin0_b = (const float*)d_in[4];
    const float* lin0_s = (const float*)d_in[5];
    const float* lin0_t = (const float*)d_in[6];
    const float* lin1_w = (const float*)d_in[7];
    const float* lin1_b = (const float*)d_in[8];
    const float* lin1_s = (const float*)d_in[9];
    const float* lin1_t = (const float*)d_in[10];
    const float* lin2_w = (const float*)d_in[11];
    const float* lin2_b = (const float*)d_in[12];
    const float* lin2_s = (const float*)d_in[13];
    const float* lin2_t = (const float*)d_in[14];
    const float* conv1_w = (const float*)d_in[15];
    const float* conv1_b = (const float*)d_in[16];
    const float* conv1_s = (const float*)d_in[17];
    const float* conv1_t = (const float*)d_in[18];
    const float* conv2_w = (const float*)d_in[19];
    const float* conv2_b = (const float*)d_in[20];
    const float* conv2_s = (const float*)d_in[21];
    const float* conv2_t = (const float*)d_in[22];

    const int* src = eidx;            // edge_index[0]
    const int* dst = eidx + NEDGES;   // edge_index[1]

    // d_out layout: [0..63] out | [64 .. 64+N*8) Z | [64+N*8 ..) x2 [N,64]
    float* out_g = (float*)d_out;
    float* Zacc  = out_g + NGRAPH * DTGT;
    float* x2    = Zacc + (size_t)NNODES * DTGT;

    // workspace layout (floats)
    float* ws = (float*)d_ws;
    float*    agg = ws;                                       // N*128
    float*    z1  = ws + (size_t)NNODES * DFEAT;              // N*8
    float*    z2  = z1 + (size_t)NNODES * DTGT;               // N*8
    float*    x1  = z2 + (size_t)NNODES * DTGT;               // N*128
    unsigned* sm  = (unsigned*)(x1 + (size_t)NNODES * DFEAT); // 224 words
    unsigned* colmax1 = sm;
    unsigned* colmax2 = sm + 8;
    float*    colsum1 = (float*)(sm + 16);
    float*    colsum2 = (float*)(sm + 24);
    unsigned* seg     = sm + 32;      // 3 x 64
    unsigned* seg0 = seg, *seg1 = seg + 64, *seg2 = seg + 128;

    const int BT = 256;
    const int nZ  = NNODES * DTGT;                 // 400000
    const int bZ  = (nZ + BT - 1) / BT;
    const int nAgg = NNODES * DFEAT;               // 6.4M
    const int bAgg = (nAgg + BT - 1) / BT;
    const int nScat = NEDGES * (DFEAT / 4);        // 19.2M
    const int bScat = (nScat + BT - 1) / BT;

    constexpr int MT = 5;                          // 3125 M-tiles = 5 * 625
    // 0) init accumulators
    init_small<<<1, 256, 0, stream>>>(sm);
    zero_f<<<bAgg, BT, 0, stream>>>(agg, nAgg);

    // 1) layer-0 score head: z0 -> Z region of d_out, segmax0
    lin_head<<<bZ, BT, 0, stream>>>(x, DFEAT, lin0_w, lin0_b, lin0_s, lin0_t,
                                    batch, Zacc, nullptr, seg0);

    // 2) agg1 = segment_sum(x[src], dst)
    scatter_add<<<bScat, BT, 0, stream>>>(x, src, dst, agg, NEDGES, DFEAT);

    // 3) x1 = elu(BN((x+agg1) @ conv1_w))       [N,128] via WMMA
    {
        int waves = ((NNODES / 16) / MT) * (HID1 / 16);   // 625*8 = 5000
        int thr = waves * 32;
        conv_wmma<DFEAT, HID1, MT><<<(thr + BT - 1) / BT, BT, 0, stream>>>(
            x, agg, conv1_w, conv1_b, conv1_s, conv1_t, x1);
    }

    // 4) z1 head + colmax1 + segmax1
    lin_head<<<bZ, BT, 0, stream>>>(x1, HID1, lin1_w, lin1_b, lin1_s, lin1_t,
                                    batch, z1, colmax1, seg1);

    // 5) agg2 = segment_sum(x1[src], dst)  (reuse agg buffer)
    zero_f<<<bAgg, BT, 0, stream>>>(agg, nAgg);
    scatter_add<<<bScat, BT, 0, stream>>>(x1, src, dst, agg, NEDGES, HID1);

    // 6) x2 = elu(BN((x1+agg2) @ conv2_w))      [N,64] via WMMA, straight to d_out
    {
        int waves = ((NNODES / 16) / MT) * (HID2 / 16);   // 625*4 = 2500
        int thr = waves * 32;
        conv_wmma<HID1, HID2, MT><<<(thr + BT - 1) / BT, BT, 0, stream>>>(
            x1, agg, conv2_w, conv2_b, conv2_s, conv2_t, x2);
    }

    // 7) z2 head + colmax2 + segmax2
    lin_head<<<bZ, BT, 0, stream>>>(x2, HID2, lin2_w, lin2_b, lin2_s, lin2_t,
                                    batch, z2, colmax2, seg2);

    // 8) column softmax denominators
    col_expsum<<<bZ, BT, 0, stream>>>(z1, colmax1, colsum1);
    col_expsum<<<bZ, BT, 0, stream>>>(z2, colmax2, colsum2);

    // 9) Z += softmax(z1)*z1 + softmax(z2)*z2
    z_update<<<bZ, BT, 0, stream>>>(Zacc, z1, colmax1, colsum1, z2, colmax2, colsum2);

    // 10) out = segmax0 + segmax1 + segmax2
    out_finalize<<<1, 64, 0, stream>>>(seg, out_g);
}